// wloss_72009421685303
// MI455X (gfx1250) — compile-verified
//
#include <hip/hip_runtime.h>
#include <hip/hip_bf16.h>

// ---------------------------------------------------------------------------
// Sinkhorn OT loss (B=512, N=1000, reg=0.1, <=50 iters) for gfx1250 (CDNA5).
// Linear-domain Sinkhorn: u = a_n/(K v), v = b/(K^T u); f = reg*log(u).
// GEMMs: V_WMMA_F32_16X16X4_F32. Operands staged by the Tensor Data Mover
// (tensor_load_to_lds) with double-buffered LDS and s_wait_tensorcnt.
// B-operand is staged from the TRANSPOSED weight matrix ([n][k] order) so
// each WMMA B-fragment is a contiguous, 8B-aligned LDS pair (ds_load_b64,
// no packing movs). All loop kernels are guarded by a device convergence
// flag + max_iter so the fixed 50-iteration graph matches jax.lax.while_loop
// semantics without host synchronization.
// ---------------------------------------------------------------------------

#define REGP      0.1f
#define ERR_BOUND 1e-08f
#define BROWS     512
#define NREAL     1000
#define NP        1024      // padded N (multiple of 16/128)
#define MAXIT     50
#define KC        64        // K-chunk staged in LDS
#define XS_STRIDE 65        // 64 + 1 pad  (A tile rows;     TDM pad_interval=5, amount=0)
#define WS_STRIDE 68        // 64 + 4 pad  (B tile n-rows;   TDM pad_interval=5, amount=3)

typedef float v2f __attribute__((ext_vector_type(2)));
typedef float v8f __attribute__((ext_vector_type(8)));
typedef unsigned int v4u __attribute__((ext_vector_type(4)));
typedef int v4i __attribute__((ext_vector_type(4)));
typedef int v8i __attribute__((ext_vector_type(8)));

// TDM group1 dword0: data_size=4B (2<<16), pad_enable (1<<20), pad_interval, pad_amount
#define XPADCFG ((2u << 16) | (1u << 20) | (5u << 22) | (0u << 25))  // 64 DW rows + 1 DW pad
#define WPADCFG ((2u << 16) | (1u << 20) | (5u << 22) | (3u << 25))  // 64 DW rows + 4 DW pad

// ---------------------------------------------------------------------------
// Issue one 2D TDM tile load into LDS at lds_addr: a tile_d0 x tile_d1 tile at
// gaddr inside a tensor of (tensor_d0 x tensor_d1), row stride stride0
// elements. OOB rows/cols (beyond tensor dims, measured from the tile start)
// are zero-filled by the TDM. Tracked by TENSORcnt.
// ---------------------------------------------------------------------------
__device__ __forceinline__ void tdm_load_2d(unsigned lds_addr, const float* gaddr,
                                            unsigned tensor_d0, unsigned tensor_d1,
                                            unsigned tile_d0, unsigned tile_d1,
                                            unsigned stride0, unsigned padcfg)
{
    const unsigned long long ga = (unsigned long long)gaddr;
    v4u g0;
    g0[0] = 1u;                                               // count=1, user descriptor
    g0[1] = lds_addr;                                         // lds_addr[31:0]
    g0[2] = (unsigned)(ga & 0xFFFFFFFFu);                     // global_addr[31:0]
    g0[3] = (unsigned)((ga >> 32) & 0x01FFFFFFu) | (2u << 30);// global_addr[56:32] | type=2
    v8i g1;
    g1[0] = (int)padcfg;                                      // data_size / pad config
    g1[1] = (int)((tensor_d0 & 0xFFFFu) << 16);               // [63:48]=tensor_dim0[15:0]
    g1[2] = (int)((tensor_d0 >> 16) | ((tensor_d1 & 0xFFFFu) << 16));
    g1[3] = (int)((tensor_d1 >> 16) | ((tile_d0 & 0xFFFFu) << 16));
    g1[4] = (int)(tile_d1 & 0xFFFFu);                         // tile_dim1 (tile_dim2=0)
    g1[5] = (int)stride0;                                     // tensor_dim0_stride[31:0]
    g1[6] = 0;
    g1[7] = 0;
    v4i z4 = {0, 0, 0, 0};
#if __clang_major__ >= 23
    v8i z8 = {0, 0, 0, 0, 0, 0, 0, 0};
    __builtin_amdgcn_tensor_load_to_lds(g0, g1, z4, z4, z8, 0);
#else
    __builtin_amdgcn_tensor_load_to_lds(g0, g1, z4, z4, 0);
#endif
}

// ---------------------------------------------------------------------------
// WMMA GEMM:  Out[512][1024] = X[512][1024](zero-padded cols) * WT^T
// WT is the TRANSPOSE of the logical weight, row-major [1000][1000]:
//   logical W[k][n] == WT[n][k];  B tiles are staged row-major from WT so
//   B-fragments {W[ka][n], W[ka+1][n]} are contiguous in LDS.
// Block: 256 threads = 8 waves; WG tile 64(M) x 128(N); wave tile 16 x 64.
// ---------------------------------------------------------------------------
__global__ __launch_bounds__(256)
void sink_gemm(const float* __restrict__ X, const float* __restrict__ WT,
               float* __restrict__ Out, const int* __restrict__ flag,
               const int* __restrict__ max_iter, int iter)
{
    if (iter >= *max_iter) return;        // while_loop: i < max_iter
    if (iter > 0 && *flag) return;        // while_loop: i==0 || err >= ep

    __shared__ float Xs[2][64 * XS_STRIDE];   // [m(64)][k(64)+pad]
    __shared__ float Ws[2][128 * WS_STRIDE];  // [n(128)][k(64)+pad]

    const int tid  = threadIdx.x;
    const int lane = tid & 31;
    const int l15  = lane & 15;
    const int kh   = lane >> 4;           // half-wave select
    const int wave = tid >> 5;
    const int wm   = wave >> 1;           // 0..3 : M sub-tile
    const int wn   = wave & 1;            // 0..1 : N sub-tile
    const int mblk = blockIdx.y * 64;
    const int nblk = blockIdx.x * 128;
    const bool issuer = (tid < 32);       // wave 0 drives the TDM

    v8f acc[4] = {};

    // Prime buffer 0
    if (issuer) {
        tdm_load_2d((unsigned)(unsigned long long)&Xs[0][0],
                    X + (unsigned)mblk * NP,
                    NP, (unsigned)(BROWS - mblk), KC, 64, NP, XPADCFG);
        tdm_load_2d((unsigned)(unsigned long long)&Ws[0][0],
                    WT + (unsigned)nblk * NREAL,
                    NREAL, (unsigned)(NREAL - nblk), KC, 128, NREAL, WPADCFG);
    }

    int buf = 0;
    for (int kc = 0; kc < NP; kc += KC) {
        if (issuer) __builtin_amdgcn_s_wait_tensorcnt(0);
        __syncthreads();                          // tiles for `buf` are visible
        const int kn = kc + KC;
        if (issuer && kn < NP) {                  // prefetch next chunk into buf^1
            tdm_load_2d((unsigned)(unsigned long long)&Xs[buf ^ 1][0],
                        X + (unsigned)mblk * NP + (unsigned)kn,
                        (unsigned)(NP - kn), (unsigned)(BROWS - mblk),
                        KC, 64, NP, XPADCFG);
            // k beyond NREAL (and n beyond NREAL-nblk) zero-filled by TDM OOB
            tdm_load_2d((unsigned)(unsigned long long)&Ws[buf ^ 1][0],
                        WT + (unsigned)nblk * NREAL + (unsigned)kn,
                        (unsigned)(kn < NREAL ? NREAL - kn : 0),
                        (unsigned)(NREAL - nblk),
                        KC, 128, NREAL, WPADCFG);
        }

        // ---- compute: 16 k-steps x 4 WMMA (A fragment reused 4x) ----
        const float* xrow = &Xs[buf][(wm * 16 + l15) * XS_STRIDE];
        const float* wbuf = &Ws[buf][0];
        #pragma unroll
        for (int kk = 0; kk < KC; kk += 4) {
            const int ka = kk + 2 * kh;            // ISA f32 A/B layout: vgpr j <-> k = 2*kh + j
            v2f afrag;
            afrag.x = xrow[ka + 0];
            afrag.y = xrow[ka + 1];
            #pragma unroll
            for (int jj = 0; jj < 4; ++jj) {
                const int nn = wn * 64 + jj * 16 + l15;
                const float* wp = wbuf + nn * WS_STRIDE + ka;   // 8B-aligned pair
                v2f bfrag;
                bfrag.x = wp[0];
                bfrag.y = wp[1];
                acc[jj] = __builtin_amdgcn_wmma_f32_16x16x4_f32(
                    false, afrag, false, bfrag, (short)0, acc[jj], false, false);
            }
        }
        __syncthreads();                          // all reads of `buf` done before reuse
        buf ^= 1;
    }

    // ---- store D: vgpr r <-> row r + 8*kh, lane l15 <-> col ----
    const int row0 = mblk + wm * 16 + kh * 8;
    #pragma unroll
    for (int jj = 0; jj < 4; ++jj) {
        const int col = nblk + wn * 64 + jj * 16 + l15;
        #pragma unroll
        for (int r = 0; r < 8; ++r)
            Out[(row0 + r) * NP + col] = acc[jj][r];
    }
}

// ---------------------------------------------------------------------------
// One-time: KT = K^T  (tiled transpose with guards, 1000x1000)
// ---------------------------------------------------------------------------
__global__ __launch_bounds__(256)
void transpose_k(const float* __restrict__ Kg, float* __restrict__ KT)
{
    __shared__ float t[32][33];
    const int bx = blockIdx.x * 32, by = blockIdx.y * 32;
    int x = bx + threadIdx.x;
    #pragma unroll
    for (int j = 0; j < 32; j += 8) {
        int y = by + threadIdx.y + j;
        if (x < NREAL && y < NREAL) t[threadIdx.y + j][threadIdx.x] = Kg[y * NREAL + x];
    }
    __syncthreads();
    x = by + threadIdx.x;
    #pragma unroll
    for (int j = 0; j < 32; j += 8) {
        int y = bx + threadIdx.y + j;
        if (x < NREAL && y < NREAL) KT[y * NREAL + x] = t[threadIdx.x][threadIdx.y + j];
    }
}

// ---------------------------------------------------------------------------
// Row reduce: count[r] = sum(a[r,:]);  u[r,:] = 1 (n<NREAL) else 0
// ---------------------------------------------------------------------------
__global__ __launch_bounds__(256)
void count_and_init_u(const float* __restrict__ a, float* __restrict__ count,
                      float* __restrict__ u)
{
    __shared__ float sm[256];
    const int r = blockIdx.x;
    float s = 0.0f;
    for (int n = threadIdx.x; n < NREAL; n += 256) s += a[r * NREAL + n];
    sm[threadIdx.x] = s;
    __syncthreads();
    for (int st = 128; st > 0; st >>= 1) {
        if (threadIdx.x < st) sm[threadIdx.x] += sm[threadIdx.x + st];
        __syncthreads();
    }
    if (threadIdx.x == 0) count[r] = sm[0];
    for (int n = threadIdx.x; n < NP; n += 256)
        u[r * NP + n] = (n < NREAL) ? 1.0f : 0.0f;
}

__global__ void init_scalars(int* flag) { *flag = 0; }

// v = b / M1   (pad cols -> 0)
__global__ __launch_bounds__(256)
void ew_v(const float* __restrict__ b, const float* __restrict__ M1,
          float* __restrict__ v, const int* flag, const int* max_iter, int iter)
{
    if (iter >= *max_iter) return;
    if (iter > 0 && *flag) return;
    const int idx = blockIdx.x * 256 + threadIdx.x;
    const int n = idx & (NP - 1);
    const int r = idx >> 10;
    v[idx] = (n < NREAL) ? b[r * NREAL + n] / M1[idx] : 0.0f;
}

// u = (a/count) / M2   (pad cols -> 0)
__global__ __launch_bounds__(256)
void ew_u(const float* __restrict__ a, const float* __restrict__ count,
          const float* __restrict__ M2, float* __restrict__ u,
          const int* flag, const int* max_iter, int iter)
{
    if (iter >= *max_iter) return;
    if (iter > 0 && *flag) return;
    const int idx = blockIdx.x * 256 + threadIdx.x;
    const int n = idx & (NP - 1);
    const int r = idx >> 10;
    u[idx] = (n < NREAL) ? (a[r * NREAL + n] / count[r]) / M2[idx] : 0.0f;
}

// rowred[r] = || v[r,:]*M1[r,:] - b[r,:] ||_2   (M1 = u_new @ K)
__global__ __launch_bounds__(256)
void err_row(const float* __restrict__ v, const float* __restrict__ M1,
             const float* __restrict__ b, float* __restrict__ rowred,
             const int* flag, const int* max_iter, int iter)
{
    if (iter >= *max_iter) return;
    if (iter > 0 && *flag) return;
    __shared__ float sm[256];
    const int r = blockIdx.x;
    float s = 0.0f;
    for (int n = threadIdx.x; n < NREAL; n += 256) {
        float d = v[r * NP + n] * M1[r * NP + n] - b[r * NREAL + n];
        s += d * d;
    }
    sm[threadIdx.x] = s;
    __syncthreads();
    for (int st = 128; st > 0; st >>= 1) {
        if (threadIdx.x < st) sm[threadIdx.x] += sm[threadIdx.x + st];
        __syncthreads();
    }
    if (threadIdx.x == 0) rowred[r] = sqrtf(sm[0]);
}

// err = sum(rowred); if err < ep -> flag = 1
__global__ __launch_bounds__(256)
void err_fin(const float* __restrict__ rowred, int* flag,
             const int* max_iter, int iter)
{
    if (iter >= *max_iter) return;
    if (iter > 0 && *flag) return;
    __shared__ float sm[256];
    sm[threadIdx.x] = rowred[threadIdx.x] + rowred[threadIdx.x + 256];
    __syncthreads();
    for (int st = 128; st > 0; st >>= 1) {
        if (threadIdx.x < st) sm[threadIdx.x] += sm[threadIdx.x + st];
        __syncthreads();
    }
    if (threadIdx.x == 0 && sm[0] < ERR_BOUND) *flag = 1;
}

// saf[r] = sum(a[r,:] * f[r,:]),  f = reg*log(u)
__global__ __launch_bounds__(256)
void saf_row(const float* __restrict__ a, const float* __restrict__ u,
             float* __restrict__ saf)
{
    __shared__ float sm[256];
    const int r = blockIdx.x;
    float s = 0.0f;
    for (int n = threadIdx.x; n < NREAL; n += 256)
        s += a[r * NREAL + n] * (REGP * __logf(u[r * NP + n]));
    sm[threadIdx.x] = s;
    __syncthreads();
    for (int st = 128; st > 0; st >>= 1) {
        if (threadIdx.x < st) sm[threadIdx.x] += sm[threadIdx.x + st];
        __syncthreads();
    }
    if (threadIdx.x == 0) saf[r] = sm[0];
}

// P[r] = sum_n a * ( f/count - saf/count^2 )
__global__ __launch_bounds__(256)
void loss_row(const float* __restrict__ a, const float* __restrict__ u,
              const float* __restrict__ count, const float* __restrict__ saf,
              float* __restrict__ P)
{
    __shared__ float sm[256];
    const int r = blockIdx.x;
    const float c = count[r];
    const float g2 = saf[r] / (c * c);
    float s = 0.0f;
    for (int n = threadIdx.x; n < NREAL; n += 256) {
        float f = REGP * __logf(u[r * NP + n]);
        s += a[r * NREAL + n] * (f / c - g2);
    }
    sm[threadIdx.x] = s;
    __syncthreads();
    for (int st = 128; st > 0; st >>= 1) {
        if (threadIdx.x < st) sm[threadIdx.x] += sm[threadIdx.x + st];
        __syncthreads();
    }
    if (threadIdx.x == 0) P[r] = sm[0];
}

__global__ __launch_bounds__(256)
void loss_fin(const float* __restrict__ P, float* __restrict__ out)
{
    __shared__ float sm[256];
    sm[threadIdx.x] = P[threadIdx.x] + P[threadIdx.x + 256];
    __syncthreads();
    for (int st = 128; st > 0; st >>= 1) {
        if (threadIdx.x < st) sm[threadIdx.x] += sm[threadIdx.x + st];
        __syncthreads();
    }
    if (threadIdx.x == 0) out[0] = sm[0];
}

// ---------------------------------------------------------------------------
extern "C" void kernel_launch(void* const* d_in, const int* in_sizes, int n_in,
                              void* d_out, int out_size, void* d_ws, size_t ws_size,
                              hipStream_t stream)
{
    const float* a  = (const float*)d_in[0];   // [512,1000]
    const float* b  = (const float*)d_in[1];   // [512,1000]
    const float* Kg = (const float*)d_in[2];   // [1000,1000]
    const int*   mi = (const int*)d_in[3];     // max_iter (device scalar)
    float* out = (float*)d_out;

    float* ws     = (float*)d_ws;
    float* u      = ws;                         // [512,1024]
    float* v      = u  + BROWS * NP;            // [512,1024]
    float* M1     = v  + BROWS * NP;            // [512,1024]
    float* M2     = M1 + BROWS * NP;            // [512,1024]
    float* count  = M2 + BROWS * NP;            // [512]
    float* saf    = count + BROWS;              // [512]
    float* rowred = saf + BROWS;                // [512]
    int*   flag   = (int*)(rowred + BROWS);     // convergence flag
    float* KT     = (float*)(flag + 4);         // [1000,1000], 16B aligned

    const dim3 gblk(256);
    const dim3 ggrid(NP / 128, BROWS / 64);     // (8, 8)
    const dim3 tgrid(32, 32);                   // transpose tiles
    const dim3 tblk(32, 8);
    const int  ewBlocks = (BROWS * NP) / 256;   // 2048

    count_and_init_u<<<BROWS, gblk, 0, stream>>>(a, count, u);
    init_scalars<<<1, 1, 0, stream>>>(flag);
    transpose_k<<<tgrid, tblk, 0, stream>>>(Kg, KT);

    // M1 = u0 @ K :  B-tiles staged from KT (= transpose of logical weight K)
    sink_gemm<<<ggrid, gblk, 0, stream>>>(u, KT, M1, flag, mi, 0);

    for (int i = 0; i < MAXIT; ++i) {
        ew_v<<<ewBlocks, gblk, 0, stream>>>(b, M1, v, flag, mi, i);
        // M2 = v @ K^T : B-tiles staged from K (= transpose of logical weight K^T)
        sink_gemm<<<ggrid, gblk, 0, stream>>>(v, Kg, M2, flag, mi, i);
        ew_u<<<ewBlocks, gblk, 0, stream>>>(a, count, M2, u, flag, mi, i);
        // M1 = u @ K  (also next iteration's M1)
        sink_gemm<<<ggrid, gblk, 0, stream>>>(u, KT, M1, flag, mi, i);
        err_row<<<BROWS, gblk, 0, stream>>>(v, M1, b, rowred, flag, mi, i);
        err_fin<<<1, gblk, 0, stream>>>(rowred, flag, mi, i);
    }

    saf_row<<<BROWS, gblk, 0, stream>>>(a, u, saf);
    loss_row<<<BROWS, gblk, 0, stream>>>(a, u, count, saf, rowred);
    loss_fin<<<1, gblk, 0, stream>>>(rowred, out);
}